// AutoregressiveGraphGenerator_17712445129056
// MI455X (gfx1250) — compile-verified
//
#include <hip/hip_runtime.h>
#include <hip/hip_bf16.h>
#include <math.h>

typedef float v2f __attribute__((ext_vector_type(2)));
typedef float v8f __attribute__((ext_vector_type(8)));

__device__ __forceinline__ float gelu_exact(float x) {
  return 0.5f * x * (1.0f + erff(x * 0.70710678118654752440f));
}

// ---------------------------------------------------------------------------
// LDS-staged WMMA fp32 GEMM: C = epi(A @ B + bias).
// A (M,K) row-major, B (K,N) row-major. Block = 256 threads = 8 waves in a
// 4x2 grid; block tile 64x32, K-chunk 32. Each wave computes one 16x16 tile
// with V_WMMA_F32_16X16X4_F32 (8 WMMAs per K-chunk, fragments from LDS).
// Requires: M % 64 == 0, N % 32 == 0, K % 32 == 0 (true for all call sites).
// EPI: 0 = none, 1 = relu, 2 = gelu. bias may be nullptr.
// ---------------------------------------------------------------------------
template <int EPI>
__global__ __launch_bounds__(256) void gemm_wmma_kernel(
    const float* __restrict__ A, const float* __restrict__ B,
    const float* __restrict__ bias, float* __restrict__ C,
    int M, int N, int K) {
  __shared__ float As[64][34];   // [m][k], stride 34 keeps b64 alignment
  __shared__ float Bt[32][34];   // [n][k] (transposed on fill)
  const int tid = threadIdx.x;
  const int lane = tid & 31;
  const int w = tid >> 5;
  const int wm = w & 3;          // wave row (4 in M)
  const int wn = w >> 2;         // wave col (2 in N)
  const int bn_count = N >> 5;
  const int bm = blockIdx.x / bn_count;
  const int bn = blockIdx.x % bn_count;

  const int mloc = wm * 16 + (lane & 15);  // A frag: M = lane&15 (both halves)
  const int nloc = wn * 16 + (lane & 15);  // B/C frag: N striped across lanes
  const int klo = (lane >> 4) << 1;        // lanes 16-31 hold K=2,3 of each 4

  v8f acc;
#pragma unroll
  for (int i = 0; i < 8; ++i) acc[i] = 0.0f;

  const int a_row = tid >> 3;              // A fill: 2 float4 / thread
  const int a_c4  = (tid & 7) << 2;
  const int b_row = tid >> 3;              // B fill: 1 float4 / thread
  const int b_c4  = (tid & 7) << 2;

  for (int k0 = 0; k0 < K; k0 += 32) {
    // stage A 64x32 (coalesced b128 loads)
    {
      const float4 va0 = *(const float4*)(A + (size_t)(bm * 64 + a_row) * K + k0 + a_c4);
      const float4 va1 = *(const float4*)(A + (size_t)(bm * 64 + a_row + 32) * K + k0 + a_c4);
      As[a_row][a_c4 + 0] = va0.x; As[a_row][a_c4 + 1] = va0.y;
      As[a_row][a_c4 + 2] = va0.z; As[a_row][a_c4 + 3] = va0.w;
      As[a_row + 32][a_c4 + 0] = va1.x; As[a_row + 32][a_c4 + 1] = va1.y;
      As[a_row + 32][a_c4 + 2] = va1.z; As[a_row + 32][a_c4 + 3] = va1.w;
    }
    // stage B 32x32 transposed -> Bt[n][k]
    {
      const float4 vb = *(const float4*)(B + (size_t)(k0 + b_row) * N + bn * 32 + b_c4);
      Bt[b_c4 + 0][b_row] = vb.x; Bt[b_c4 + 1][b_row] = vb.y;
      Bt[b_c4 + 2][b_row] = vb.z; Bt[b_c4 + 3][b_row] = vb.w;
    }
    // prefetch next K-chunk while we compute this one
    if (k0 + 32 < K) {
      __builtin_prefetch(A + (size_t)(bm * 64 + a_row) * K + k0 + 32 + a_c4, 0, 0);
      __builtin_prefetch(B + (size_t)(k0 + 32 + b_row) * N + bn * 32 + b_c4, 0, 0);
    }
    __syncthreads();
#pragma unroll
    for (int kk = 0; kk < 32; kk += 4) {
      v2f a, b;
      a.x = As[mloc][kk + klo];
      a.y = As[mloc][kk + klo + 1];
      b.x = Bt[nloc][kk + klo];
      b.y = Bt[nloc][kk + klo + 1];
      acc = __builtin_amdgcn_wmma_f32_16x16x4_f32(false, a, false, b, (short)0,
                                                  acc, false, false);
    }
    __syncthreads();
  }
  const int col = bn * 32 + nloc;
  const int mbase = bm * 64 + wm * 16 + ((lane >> 4) << 3);  // C: hi lanes M=v+8
  const float bv = bias ? bias[col] : 0.0f;
#pragma unroll
  for (int v = 0; v < 8; ++v) {
    float x = acc[v] + bv;
    if (EPI == 1) x = fmaxf(x, 0.0f);
    if (EPI == 2) x = gelu_exact(x);
    C[(size_t)(mbase + v) * N + col] = x;
  }
}

static inline void launch_gemm(int epi, const float* A, const float* B,
                               const float* bias, float* C, int M, int N, int K,
                               hipStream_t s) {
  int blocks = (M >> 6) * (N >> 5);
  switch (epi) {
    case 0: gemm_wmma_kernel<0><<<blocks, 256, 0, s>>>(A, B, bias, C, M, N, K); break;
    case 1: gemm_wmma_kernel<1><<<blocks, 256, 0, s>>>(A, B, bias, C, M, N, K); break;
    default: gemm_wmma_kernel<2><<<blocks, 256, 0, s>>>(A, B, bias, C, M, N, K); break;
  }
}

// ---------------------------------------------------------------------------
// out[row] = LN(x[row] + res[row or row/64]) * g + beta   (H = 256 fixed)
// ---------------------------------------------------------------------------
__global__ __launch_bounds__(256) void add_ln_kernel(
    const float* __restrict__ x, const float* __restrict__ res,
    const float* __restrict__ g, const float* __restrict__ beta,
    float* __restrict__ out, int bcast) {
  __shared__ float red[256];
  const int row = blockIdx.x, tid = threadIdx.x;
  float v = x[(size_t)row * 256 + tid];
  if (res) {
    int rr = bcast ? (row >> 6) : row;
    v += res[(size_t)rr * 256 + tid];
  }
  red[tid] = v; __syncthreads();
  for (int s = 128; s > 0; s >>= 1) { if (tid < s) red[tid] += red[tid + s]; __syncthreads(); }
  const float mu = red[0] * (1.0f / 256.0f);
  __syncthreads();
  const float d = v - mu;
  red[tid] = d * d; __syncthreads();
  for (int s = 128; s > 0; s >>= 1) { if (tid < s) red[tid] += red[tid + s]; __syncthreads(); }
  const float var = red[0] * (1.0f / 256.0f);
  out[(size_t)row * 256 + tid] = d * rsqrtf(var + 1e-5f) * g[tid] + beta[tid];
}

// memory[b] = LN(z[b] @ lp_w + lp_b) * lp_g + lp_beta   (L=128 -> H=256)
__global__ __launch_bounds__(256) void latent_kernel(
    const float* __restrict__ z, const float* __restrict__ w,
    const float* __restrict__ b, const float* __restrict__ g,
    const float* __restrict__ beta, float* __restrict__ mem) {
  __shared__ float red[256];
  const int bb = blockIdx.x, tid = threadIdx.x;
  const float* zr = z + bb * 128;
  float acc = b[tid];
  for (int k = 0; k < 128; ++k) acc += zr[k] * w[k * 256 + tid];
  red[tid] = acc; __syncthreads();
  for (int s = 128; s > 0; s >>= 1) { if (tid < s) red[tid] += red[tid + s]; __syncthreads(); }
  const float mu = red[0] * (1.0f / 256.0f);
  __syncthreads();
  const float d = acc - mu;
  red[tid] = d * d; __syncthreads();
  for (int s = 128; s > 0; s >>= 1) { if (tid < s) red[tid] += red[tid + s]; __syncthreads(); }
  const float var = red[0] * (1.0f / 256.0f);
  mem[(size_t)bb * 256 + tid] = d * rsqrtf(var + 1e-5f) * g[tid] + beta[tid];
}

// seq[b,n,:] = tok_emb[shifted_tok] + pos_emb[n]
__global__ __launch_bounds__(256) void embed_kernel(
    const int* __restrict__ toks, const float* __restrict__ tok_emb,
    const float* __restrict__ pos_emb, float* __restrict__ seq) {
  const int idx = blockIdx.x * 256 + threadIdx.x;  // B*N*H threads
  const int c = idx & 255, row = idx >> 8, n = row & 63, b = row >> 6;
  const int tok = (n == 0) ? 0 : toks[b * 64 + n - 1];
  seq[idx] = tok_emb[tok * 256 + c] + pos_emb[n * 256 + c];
}

// Causal self-attention, one block per (b,h). N=64, hd=64.
__global__ __launch_bounds__(256) void attn_kernel(
    const float* __restrict__ q, const float* __restrict__ k,
    const float* __restrict__ v, float* __restrict__ out) {
  __shared__ float Qs[4096];  // reused as scores after phase 1
  __shared__ float Ks[4096];
  __shared__ float Vs[4096];
  const int h = blockIdx.x & 3, b = blockIdx.x >> 2, tid = threadIdx.x;
  const size_t base = (size_t)b * 64 * 256 + h * 64;
  for (int i = tid; i < 4096; i += 256) {
    const int n = i >> 6, d = i & 63;
    Qs[i] = q[base + (size_t)n * 256 + d] * 0.125f;  // 1/sqrt(64)
    Ks[i] = k[base + (size_t)n * 256 + d];
    Vs[i] = v[base + (size_t)n * 256 + d];
  }
  __syncthreads();
  float sc[16];
#pragma unroll 1
  for (int r = 0; r < 16; ++r) {
    const int s = r * 256 + tid, qi = s >> 6, kj = s & 63;
    float acc = 0.0f;
    for (int d = 0; d < 64; ++d) acc += Qs[qi * 64 + d] * Ks[kj * 64 + d];
    sc[r] = acc;
  }
  __syncthreads();
  for (int r = 0; r < 16; ++r) Qs[r * 256 + tid] = sc[r];
  __syncthreads();
  if (tid < 64) {  // causal softmax, one thread per query row
    float* rowp = Qs + tid * 64;
    float mx = -1e30f;
    for (int j = 0; j <= tid; ++j) mx = fmaxf(mx, rowp[j]);
    float sum = 0.0f;
    for (int j = 0; j <= tid; ++j) { float e = expf(rowp[j] - mx); rowp[j] = e; sum += e; }
    const float inv = 1.0f / sum;
    for (int j = 0; j <= tid; ++j) rowp[j] *= inv;
    for (int j = tid + 1; j < 64; ++j) rowp[j] = 0.0f;
  }
  __syncthreads();
  for (int i = tid; i < 4096; i += 256) {
    const int qi = i >> 6, d = i & 63;
    float acc = 0.0f;
    for (int j = 0; j < 64; ++j) acc += Qs[qi * 64 + j] * Vs[j * 64 + d];
    out[base + (size_t)qi * 256 + d] = acc;
  }
}

// Cross-attn with memory length 1: softmax == 1, so
// cavec[b] = (mem[b] @ wv + bv) @ wo + bo.  One block per batch.
__global__ __launch_bounds__(256) void cavec_kernel(
    const float* __restrict__ mem, const float* __restrict__ wv,
    const float* __restrict__ bv, const float* __restrict__ wo,
    const float* __restrict__ bo, float* __restrict__ out) {
  __shared__ float vsh[256];
  const int b = blockIdx.x, t = threadIdx.x;
  const float* m = mem + b * 256;
  float acc = bv[t];
  for (int kk = 0; kk < 256; ++kk) acc += m[kk] * wv[kk * 256 + t];
  vsh[t] = acc; __syncthreads();
  float o = bo[t];
  for (int kk = 0; kk < 256; ++kk) o += vsh[kk] * wo[kk * 256 + t];
  out[b * 256 + t] = o;
}

// dinv[b,i] = deg>0 ? rsqrt(deg) : 0, deg = sum_j A*mi*mj + mi
__global__ __launch_bounds__(256) void dinv_kernel(
    const float* __restrict__ A, const unsigned char* __restrict__ mask,
    float* __restrict__ dinv) {
  const int idx = blockIdx.x * 256 + threadIdx.x;  // B*N
  const int b = idx >> 6, i = idx & 63;
  const float mi = mask[idx] ? 1.0f : 0.0f;
  const float* ar = A + (size_t)b * 4096 + i * 64;
  const unsigned char* mb = mask + b * 64;
  float deg = mi;
  for (int j = 0; j < 64; ++j) deg += ar[j] * mi * (mb[j] ? 1.0f : 0.0f);
  dinv[idx] = (deg > 0.0f) ? rsqrtf(fmaxf(deg, 1e-12f)) : 0.0f;
}

__global__ __launch_bounds__(256) void anorm_kernel(
    const float* __restrict__ A, const unsigned char* __restrict__ mask,
    const float* __restrict__ dinv, float* __restrict__ An) {
  const int idx = blockIdx.x * 256 + threadIdx.x;  // B*N*N
  const int j = idx & 63, i = (idx >> 6) & 63, b = idx >> 12;
  const float mi = mask[b * 64 + i] ? 1.0f : 0.0f;
  const float mj = mask[b * 64 + j] ? 1.0f : 0.0f;
  const float a = A[idx] * mi * mj + ((i == j) ? mi : 0.0f);
  An[idx] = dinv[b * 64 + i] * a * dinv[b * 64 + j];
}

// gn[b,i,:] = sum_j Anorm[b,i,j] * t[b,j,:] + bias. One block per (b,i).
__global__ __launch_bounds__(256) void spmm_kernel(
    const float* __restrict__ An, const float* __restrict__ t,
    const float* __restrict__ bias, float* __restrict__ gn) {
  __shared__ float arow[64];
  const int bi = blockIdx.x, b = bi >> 6, tid = threadIdx.x;
  if (tid < 64) arow[tid] = An[(size_t)bi * 64 + tid];
  __syncthreads();
  float acc = bias[tid];
  const float* tb = t + (size_t)b * 64 * 256 + tid;
  for (int j = 0; j < 64; ++j) acc += arow[j] * tb[j * 256];
  gn[(size_t)bi * 256 + tid] = acc;
}

__global__ __launch_bounds__(256) void geluadd_kernel(float* __restrict__ g,
                                                      const float* __restrict__ x) {
  const int idx = blockIdx.x * 256 + threadIdx.x;
  g[idx] += gelu_exact(x[idx]);
}

__global__ __launch_bounds__(256) void select_kernel(
    const float* __restrict__ g, const float* __restrict__ hid0,
    const unsigned char* __restrict__ mask, float* __restrict__ hid,
    float* __restrict__ out_hid) {
  const int idx = blockIdx.x * 256 + threadIdx.x;
  const int row = idx >> 8;
  const float v = mask[row] ? g[idx] : hid0[idx];
  hid[idx] = v;
  out_hid[idx] = v;
}

__global__ __launch_bounds__(256) void copy_kernel(float* __restrict__ dst,
                                                   const float* __restrict__ src) {
  const int idx = blockIdx.x * 256 + threadIdx.x;
  dst[idx] = src[idx];
}

// edge[b,i,j] = sum_h gelu(qa[b,i,h]+kb[b,j,h]+b1[h]) * w2[h] + b2, masked.
__global__ __launch_bounds__(256) void edge_kernel(
    const float* __restrict__ qa, const float* __restrict__ kb,
    const float* __restrict__ b1, const float* __restrict__ w2,
    const float* __restrict__ b2, const unsigned char* __restrict__ mask,
    float* __restrict__ out) {
  __shared__ float sQ[256];
  __shared__ float red[256];
  const int bi = blockIdx.x, b = bi >> 6, tid = threadIdx.x;
  sQ[tid] = qa[(size_t)bi * 256 + tid] + b1[tid];
  __syncthreads();
  const int j = tid >> 2, grp = tid & 3;
  const float* kr = kb + (size_t)(b * 64 + j) * 256 + grp * 64;
  const float* qf = sQ + grp * 64;
  const float* wf = w2 + grp * 64;
  float acc = 0.0f;
  for (int hh = 0; hh < 64; ++hh) acc += gelu_exact(qf[hh] + kr[hh]) * wf[hh];
  red[tid] = acc;
  __syncthreads();
  if (grp == 0) {
    const float s = red[tid] + red[tid + 1] + red[tid + 2] + red[tid + 3] + b2[0];
    const bool em = mask[bi] && mask[b * 64 + j];
    out[(size_t)bi * 64 + j] = em ? s : -__builtin_huge_valf();
  }
}

// stop[row] = gelu(hid[row] @ w1 + b1) @ w2 + b2. One block (128 thr) per row.
__global__ __launch_bounds__(128) void stop_kernel(
    const float* __restrict__ hid, const float* __restrict__ w1,
    const float* __restrict__ b1, const float* __restrict__ w2,
    const float* __restrict__ b2, float* __restrict__ out) {
  __shared__ float red[128];
  const int row = blockIdx.x, tid = threadIdx.x;
  const float* hr = hid + (size_t)row * 256;
  float acc = b1[tid];
  for (int k = 0; k < 256; ++k) acc += hr[k] * w1[k * 128 + tid];
  red[tid] = gelu_exact(acc) * w2[tid];
  __syncthreads();
  for (int s = 64; s > 0; s >>= 1) { if (tid < s) red[tid] += red[tid + s]; __syncthreads(); }
  if (tid == 0) out[row] = red[0] + b2[0];
}

// ---------------------------------------------------------------------------
extern "C" void kernel_launch(void* const* d_in, const int* in_sizes, int n_in,
                              void* d_out, int out_size, void* d_ws, size_t ws_size,
                              hipStream_t stream) {
  (void)in_sizes; (void)n_in; (void)out_size; (void)ws_size;
  const int ROWS = 64 * 64;        // B*N = 4096
  const int RH   = ROWS * 256;     // 1,048,576

  const float* z   = (const float*)d_in[0];
  const int* toks  = (const int*)d_in[1];
  const float* adj = (const float*)d_in[2];
  const unsigned char* mask = (const unsigned char*)d_in[3];
  const float* lp_w = (const float*)d_in[4];
  const float* lp_b = (const float*)d_in[5];
  const float* lp_g = (const float*)d_in[6];
  const float* lp_beta = (const float*)d_in[7];
  const float* tok_emb = (const float*)d_in[8];
  const float* pos_emb = (const float*)d_in[9];
  const float* sa_wq = (const float*)d_in[10];
  const float* sa_wk = (const float*)d_in[11];
  const float* sa_wv = (const float*)d_in[12];
  const float* sa_wo = (const float*)d_in[13];
  const float* ca_wv = (const float*)d_in[16];
  const float* ca_wo = (const float*)d_in[17];
  const float* ff_w1 = (const float*)d_in[18];
  const float* ff_w2 = (const float*)d_in[19];
  const float* sa_bq = (const float*)d_in[20];
  const float* sa_bk = (const float*)d_in[21];
  const float* sa_bv = (const float*)d_in[22];
  const float* sa_bo = (const float*)d_in[23];
  const float* ca_bv = (const float*)d_in[26];
  const float* ca_bo = (const float*)d_in[27];
  const float* ff_b1 = (const float*)d_in[28];
  const float* ff_b2 = (const float*)d_in[29];
  const float* ln1_b = (const float*)d_in[30];
  const float* ln2_b = (const float*)d_in[31];
  const float* ln3_b = (const float*)d_in[32];
  const float* ln1_g = (const float*)d_in[33];
  const float* ln2_g = (const float*)d_in[34];
  const float* ln3_g = (const float*)d_in[35];
  const float* gnn_w = (const float*)d_in[36];
  const float* gnn_b = (const float*)d_in[37];
  const float* gnn_g = (const float*)d_in[38];
  const float* gnn_beta = (const float*)d_in[39];
  const float* nh_w1 = (const float*)d_in[40];
  const float* nh_b1 = (const float*)d_in[41];
  const float* nh_w2 = (const float*)d_in[42];
  const float* nh_b2 = (const float*)d_in[43];
  const float* eq_w = (const float*)d_in[44];
  const float* eq_b = (const float*)d_in[45];
  const float* ek_w = (const float*)d_in[46];
  const float* ek_b = (const float*)d_in[47];
  const float* em_w1a = (const float*)d_in[48];
  const float* em_w1b = (const float*)d_in[49];
  const float* em_b1 = (const float*)d_in[50];
  const float* em_w2 = (const float*)d_in[51];
  const float* em_b2 = (const float*)d_in[52];
  const float* sh_w1 = (const float*)d_in[53];
  const float* sh_b1 = (const float*)d_in[54];
  const float* sh_w2 = (const float*)d_in[55];
  const float* sh_b2 = (const float*)d_in[56];

  // workspace layout (floats)
  float* ws = (float*)d_ws;
  float* mem   = ws;               // 16384
  float* cav   = mem + 16384;      // 16384
  float* hA    = cav + 16384;      // RH
  float* hB    = hA + RH;          // RH
  float* qb    = hB + RH;          // RH
  float* kbuf  = qb + RH;          // RH
  float* vb    = kbuf + RH;        // RH
  float* att   = vb + RH;          // RH
  float* tmp   = att + RH;         // RH
  float* mid   = tmp + RH;         // ROWS*1024
  float* anorm = mid + ROWS * 1024;// 262144
  float* dinv  = anorm + 262144;   // 4096
  float* gbuf  = dinv + 4096;      // RH
  float* hid0  = gbuf + RH;        // RH

  float* out_node = (float*)d_out;                 // B*N*T
  float* out_edge = out_node + RH;                 // B*N*N
  float* out_stop = out_edge + 262144;             // B*N
  float* out_hid  = out_stop + 4096;               // B*N*H

  // ---- latent memory + embeddings -----------------------------------------
  latent_kernel<<<64, 256, 0, stream>>>(z, lp_w, lp_b, lp_g, lp_beta, mem);
  embed_kernel<<<RH / 256, 256, 0, stream>>>(toks, tok_emb, pos_emb, hA);

  // ---- 4 transformer decoder layers ---------------------------------------
  float* cur = hA;
  float* alt = hB;
  for (int l = 0; l < 4; ++l) {
    const int wOff = l * 65536, bOff = l * 256;
    // self-attention: QKV projections (WMMA), attention, output projection
    launch_gemm(0, cur, sa_wq + wOff, sa_bq + bOff, qb,   ROWS, 256, 256, stream);
    launch_gemm(0, cur, sa_wk + wOff, sa_bk + bOff, kbuf, ROWS, 256, 256, stream);
    launch_gemm(0, cur, sa_wv + wOff, sa_bv + bOff, vb,   ROWS, 256, 256, stream);
    attn_kernel<<<256, 256, 0, stream>>>(qb, kbuf, vb, att);
    launch_gemm(0, att, sa_wo + wOff, sa_bo + bOff, tmp, ROWS, 256, 256, stream);
    add_ln_kernel<<<ROWS, 256, 0, stream>>>(tmp, cur, ln1_g + bOff, ln1_b + bOff, alt, 0);
    { float* t2 = cur; cur = alt; alt = t2; }
    // cross-attention (memory len 1 -> per-batch vector, broadcast residual)
    cavec_kernel<<<64, 256, 0, stream>>>(mem, ca_wv + wOff, ca_bv + bOff,
                                         ca_wo + wOff, ca_bo + bOff, cav);
    add_ln_kernel<<<ROWS, 256, 0, stream>>>(cur, cav, ln2_g + bOff, ln2_b + bOff, alt, 1);
    { float* t2 = cur; cur = alt; alt = t2; }
    // FFN (relu)
    launch_gemm(1, cur, ff_w1 + l * 262144, ff_b1 + l * 1024, mid, ROWS, 1024, 256, stream);
    launch_gemm(0, mid, ff_w2 + l * 262144, ff_b2 + bOff, tmp, ROWS, 256, 1024, stream);
    add_ln_kernel<<<ROWS, 256, 0, stream>>>(tmp, cur, ln3_g + bOff, ln3_b + bOff, alt, 0);
    { float* t2 = cur; cur = alt; alt = t2; }
  }
  copy_kernel<<<RH / 256, 256, 0, stream>>>(hid0, cur);
  copy_kernel<<<RH / 256, 256, 0, stream>>>(gbuf, cur);

  // ---- GCN refine ----------------------------------------------------------
  dinv_kernel<<<ROWS / 256, 256, 0, stream>>>(adj, mask, dinv);
  anorm_kernel<<<262144 / 256, 256, 0, stream>>>(adj, mask, dinv, anorm);
  for (int l = 0; l < 2; ++l) {
    launch_gemm(0, gbuf, gnn_w + l * 65536, nullptr, vb, ROWS, 256, 256, stream);
    spmm_kernel<<<ROWS, 256, 0, stream>>>(anorm, vb, gnn_b + l * 256, att);
    add_ln_kernel<<<ROWS, 256, 0, stream>>>(att, nullptr, gnn_g + l * 256,
                                            gnn_beta + l * 256, tmp, 0);
    geluadd_kernel<<<RH / 256, 256, 0, stream>>>(gbuf, tmp);
  }
  float* hid = qb;  // reuse QKV scratch
  select_kernel<<<RH / 256, 256, 0, stream>>>(gbuf, hid0, mask, hid, out_hid);

  // ---- node head: gelu(hid@W1+b1)@W2+b2 ------------------------------------
  launch_gemm(2, hid, nh_w1, nh_b1, tmp, ROWS, 256, 256, stream);
  launch_gemm(0, tmp, nh_w2, nh_b2, out_node, ROWS, 256, 256, stream);

  // ---- edge head -----------------------------------------------------------
  launch_gemm(0, hid, eq_w, eq_b, att, ROWS, 256, 256, stream);   // q
  launch_gemm(0, hid, ek_w, ek_b, vb, ROWS, 256, 256, stream);    // k
  launch_gemm(0, att, em_w1a, nullptr, kbuf, ROWS, 256, 256, stream);  // qa
  launch_gemm(0, vb, em_w1b, nullptr, mid, ROWS, 256, 256, stream);    // kb
  edge_kernel<<<ROWS, 256, 0, stream>>>(kbuf, mid, em_b1, em_w2, em_b2, mask, out_edge);

  // ---- stop head -----------------------------------------------------------
  stop_kernel<<<ROWS, 128, 0, stream>>>(hid, sh_w1, sh_b1, sh_w2, sh_b2, out_stop);
}